// LocalDenseConv1D_73658689126881
// MI455X (gfx1250) — compile-verified
//
#include <hip/hip_runtime.h>
#include <hip/hip_bf16.h>

typedef float v2f __attribute__((ext_vector_type(2)));
typedef float v8f __attribute__((ext_vector_type(8)));

#define B_     8
#define IN_C   64
#define IN_L   256
#define T_     256
#define OUT_C  64
#define OUT_L  128
#define KH     3
#define STRIDE_ 2
#define PAD_   1
#define CK     (IN_C * KH)      // 192
#define TB     64               // t-columns per block
#define KC     32               // K-chunk staged in LDS
#define LS     80               // padded LDS row stride (words): 2*80 % 64 == 32 -> conflict-free half-wave rows

// ---------------------------------------------------------------------------
// Pass 1: locally-connected contraction via fp32 WMMA, +bias, write pre-BN out
// grid = B * OUT_L * (T/TB) = 8*128*4 = 4096 blocks of 256 threads (8 waves)
// ---------------------------------------------------------------------------
__global__ __launch_bounds__(256) void conv_wmma_kernel(
    const float* __restrict__ x, const float* __restrict__ w,
    const float* __restrict__ bias, float* __restrict__ out)
{
    __shared__ float Asm[KC * LS];   // [k][o]  (10 KB)
    __shared__ float Bsm[KC * LS];   // [k][t]  (10 KB)

    const int gid = blockIdx.x;
    const int tc  = gid & 3;
    const int l   = (gid >> 2) & (OUT_L - 1);
    const int b   = gid >> 9;
    const int t0  = tc * TB;

    const int tid  = threadIdx.x;
    const int wv   = tid >> 5;
    const int lane = tid & 31;
    const int half = lane >> 4;
    const int lc   = lane & 15;

    // two 16x16 tiles per wave; both share the same t-subtile (same B frag)
    const int id0 = wv * 2;
    const int ob0 = (id0 & 3) * 16;
    const int ob1 = ((id0 + 1) & 3) * 16;
    const int tb  = (id0 >> 2) * 16;

    v8f acc0 = {0.f, 0.f, 0.f, 0.f, 0.f, 0.f, 0.f, 0.f};
    v8f acc1 = acc0;

    for (int kk = 0; kk < CK; kk += KC) {
        // stage A chunk: w[kk+k][o] at this l   (KC*OUT_C = 2048 elems)
        #pragma unroll
        for (int i = 0; i < 8; ++i) {
            int idx = i * 256 + tid;
            int k = idx >> 6, o = idx & 63;
            Asm[k * LS + o] = w[(kk + k) * (OUT_C * OUT_L) + o * OUT_L + l];
        }
        // stage B chunk: unfolded x window rows (coalesced along t)
        #pragma unroll
        for (int i = 0; i < 8; ++i) {
            int idx = i * 256 + tid;
            int k = idx >> 6, t = idx & 63;
            int ck = kk + k;
            int c  = ck / 3;
            int kh = ck - 3 * c;
            int h  = STRIDE_ * l + kh - PAD_;
            float v = 0.f;
            if (h >= 0 && h < IN_L)
                v = x[((b * IN_C + c) * IN_L + h) * T_ + t0 + t];
            Bsm[k * LS + t] = v;
        }
        __syncthreads();

        // 8 K=4 steps per chunk; A frag: VGPR0 holds K=k(+2 for hi half), VGPR1 K+1
        #pragma unroll
        for (int k4 = 0; k4 < KC; k4 += 4) {
            int r0 = (k4 + 2 * half) * LS;
            int r1 = r0 + LS;
            v2f bf = { Bsm[r0 + tb  + lc], Bsm[r1 + tb  + lc] };
            v2f a0 = { Asm[r0 + ob0 + lc], Asm[r1 + ob0 + lc] };
            v2f a1 = { Asm[r0 + ob1 + lc], Asm[r1 + ob1 + lc] };
            acc0 = __builtin_amdgcn_wmma_f32_16x16x4_f32(
                       false, a0, false, bf, (short)0, acc0, false, false);
            acc1 = __builtin_amdgcn_wmma_f32_16x16x4_f32(
                       false, a1, false, bf, (short)0, acc1, false, false);
        }
        __syncthreads();
    }

    // epilogue: C/D layout -> VGPR r: lanes0-15 M=r, lanes16-31 M=r+8; N=lane&15
    const int t = t0 + tb + lc;
    #pragma unroll
    for (int r = 0; r < 8; ++r) {
        int o0 = ob0 + r + 8 * half;
        out[((b * OUT_C + o0) * OUT_L + l) * T_ + t] = acc0[r] + bias[o0 * OUT_L + l];
        int o1 = ob1 + r + 8 * half;
        out[((b * OUT_C + o1) * OUT_L + l) * T_ + t] = acc1[r] + bias[o1 * OUT_L + l];
    }
}

// ---------------------------------------------------------------------------
// Pass 2: per-channel partial sum/sumsq (deterministic tree reduce, no atomics)
// grid = 64 channels * 32 chunks; each block reduces 8192 elems of one channel
// ---------------------------------------------------------------------------
__global__ __launch_bounds__(256) void reduce_kernel(
    const float* __restrict__ out, float* __restrict__ part)
{
    __shared__ float sh1[256];
    __shared__ float sh2[256];
    const int o   = blockIdx.x >> 5;
    const int j   = blockIdx.x & 31;
    const int tid = threadIdx.x;

    float s = 0.f, q = 0.f;
    #pragma unroll 4
    for (int i = 0; i < 32; ++i) {
        int m   = j * 8192 + i * 256 + tid;          // m in [0, 262144)
        int bb  = m >> 15;                           // batch
        int rem = m & 32767;                         // (l,t) flat
        float v = out[bb * (OUT_C * OUT_L * T_) + o * (OUT_L * T_) + rem];
        s += v;
        q += v * v;
    }
    sh1[tid] = s; sh2[tid] = q;
    __syncthreads();
    for (int stp = 128; stp > 0; stp >>= 1) {
        if (tid < stp) { sh1[tid] += sh1[tid + stp]; sh2[tid] += sh2[tid + stp]; }
        __syncthreads();
    }
    if (tid == 0) {
        part[(o * 32 + j) * 2 + 0] = sh1[0];
        part[(o * 32 + j) * 2 + 1] = sh2[0];
    }
}

// ---------------------------------------------------------------------------
// Pass 3: fold partials -> mean / inv_std per channel
// ---------------------------------------------------------------------------
__global__ void finalize_kernel(const float* __restrict__ part,
                                float* __restrict__ stats)
{
    int o = threadIdx.x;                              // 64 threads
    float s = 0.f, q = 0.f;
    for (int j = 0; j < 32; ++j) {
        s += part[(o * 32 + j) * 2 + 0];
        q += part[(o * 32 + j) * 2 + 1];
    }
    const float invN = 1.0f / 262144.0f;              // B*OUT_L*T
    float mean = s * invN;
    float var  = q * invN - mean * mean;
    stats[o]      = mean;
    stats[64 + o] = rsqrtf(var + 1e-5f);
}

// ---------------------------------------------------------------------------
// Pass 4: in-place BN affine + PReLU
// ---------------------------------------------------------------------------
__global__ __launch_bounds__(256) void bn_prelu_kernel(
    float* __restrict__ out, const float* __restrict__ stats,
    const float* __restrict__ gamma, const float* __restrict__ beta,
    const float* __restrict__ pa)
{
    int idx = blockIdx.x * 256 + threadIdx.x;
    int o   = (idx >> 15) & 63;                       // channel from [B,O,L,T] flat
    float v = out[idx];
    float y = (v - stats[o]) * stats[64 + o] * gamma[o] + beta[o];
    float a = pa[0];
    out[idx] = (y >= 0.f) ? y : a * y;
}

extern "C" void kernel_launch(void* const* d_in, const int* in_sizes, int n_in,
                              void* d_out, int out_size, void* d_ws, size_t ws_size,
                              hipStream_t stream)
{
    (void)in_sizes; (void)n_in; (void)out_size; (void)ws_size;
    const float* x     = (const float*)d_in[0];
    const float* w     = (const float*)d_in[1];
    const float* bias  = (const float*)d_in[2];
    const float* gamma = (const float*)d_in[3];
    const float* beta  = (const float*)d_in[4];
    const float* pa    = (const float*)d_in[5];
    float* out   = (float*)d_out;
    float* part  = (float*)d_ws;            // 64*32*2 floats
    float* stats = part + 4096;             // 128 floats: mean | inv_std

    conv_wmma_kernel<<<4096, 256, 0, stream>>>(x, w, bias, out);
    reduce_kernel<<<2048, 256, 0, stream>>>(out, part);
    finalize_kernel<<<1, 64, 0, stream>>>(part, stats);
    bn_prelu_kernel<<<65536, 256, 0, stream>>>(out, stats, gamma, beta, pa);
}